// PointTransformerSeg_68315749810352
// MI455X (gfx1250) — compile-verified
//
#include <hip/hip_runtime.h>
#include <cstddef>

typedef float v2f __attribute__((ext_vector_type(2)));
typedef float v8f __attribute__((ext_vector_type(8)));

// ---------------- network constants ----------------
static constexpr int BB = 2;
static constexpr int NL[5] = {8192, 2048, 512, 128, 32};
static constexpr int CL[5] = {32, 64, 128, 256, 512};
static constexpr int KL[5] = {8, 16, 16, 16, 16};
static constexpr int NBLK[5] = {2, 3, 4, 6, 3};
static constexpr int NCLS = 13;

static inline int nblocks(long long n) { return (int)((n + 255) / 256); }

// ================= WMMA f32 GEMM: Y = act(bn((bnIn?)(X) @ W) [+res]) ==========
// One wave computes one 16x16 tile with V_WMMA_F32_16X16X4_F32, K stepped by 4.
template <bool IN_BN>
__global__ __launch_bounds__(256) void k_linear_wmma(
    const float* __restrict__ X, const float* __restrict__ W, float* __restrict__ Y,
    int rows, int Cin, int Cout,
    const float* __restrict__ ing, const float* __restrict__ inb,
    const float* __restrict__ outg, const float* __restrict__ outb, int outRelu,
    const float* __restrict__ res, int tN)
{
  int wave = blockIdx.x * (blockDim.x >> 5) + (threadIdx.x >> 5);
  int lane = threadIdx.x & 31;
  int tilesX = (Cout + 15) >> 4;
  int tilesY = (rows + 15) >> 4;
  if (wave >= tilesX * tilesY) return;
  int tx = wave % tilesX, ty = wave / tilesX;
  int row0 = ty << 4, col0 = tx << 4;
  int hl = lane >> 4;       // half-wave selector
  int ll = lane & 15;
  int r = row0 + ll;        // A row held by this lane
  int c = col0 + ll;        // B col held by this lane
  v8f acc;
  for (int i = 0; i < 8; ++i) acc[i] = 0.0f;

  bool fullTile = (row0 + 16 <= rows) && (col0 + 16 <= Cout);
  int CinMain = Cin & ~3;

  if (fullTile) {
    const float* Xr = X + (size_t)r * Cin;
    const float* Wc = W + c;
    int k0 = 0;
    for (; k0 < CinMain; k0 += 4) {
      int ka = k0 + (hl << 1);
      float a0 = Xr[ka];
      float a1 = Xr[ka + 1];
      if (IN_BN) {
        a0 = fmaxf(a0 * ing[ka] + inb[ka], 0.0f);
        a1 = fmaxf(a1 * ing[ka + 1] + inb[ka + 1], 0.0f);
      }
      float b0 = Wc[(size_t)ka * Cout];
      float b1 = Wc[(size_t)(ka + 1) * Cout];
      v2f av; av.x = a0; av.y = a1;
      v2f bv; bv.x = b0; bv.y = b1;
      acc = __builtin_amdgcn_wmma_f32_16x16x4_f32(false, av, false, bv,
                                                  (short)0, acc, false, false);
    }
    if (k0 < Cin) {  // K tail (Cin not multiple of 4)
      int ka = k0 + (hl << 1);
      float a0 = (ka < Cin)     ? Xr[ka]     : 0.0f;
      float a1 = (ka + 1 < Cin) ? Xr[ka + 1] : 0.0f;
      if (IN_BN) {
        if (ka < Cin)     a0 = fmaxf(a0 * ing[ka]     + inb[ka],     0.0f);
        if (ka + 1 < Cin) a1 = fmaxf(a1 * ing[ka + 1] + inb[ka + 1], 0.0f);
      }
      float b0 = (ka < Cin)     ? Wc[(size_t)ka * Cout]       : 0.0f;
      float b1 = (ka + 1 < Cin) ? Wc[(size_t)(ka + 1) * Cout] : 0.0f;
      v2f av; av.x = a0; av.y = a1;
      v2f bv; bv.x = b0; bv.y = b1;
      acc = __builtin_amdgcn_wmma_f32_16x16x4_f32(false, av, false, bv,
                                                  (short)0, acc, false, false);
    }
  } else {
    bool rOK = r < rows;
    bool cOK = c < Cout;
    for (int k0 = 0; k0 < Cin; k0 += 4) {
      int ka = k0 + (hl << 1);
      float a0 = (rOK && ka < Cin)     ? X[(size_t)r * Cin + ka]     : 0.0f;
      float a1 = (rOK && ka + 1 < Cin) ? X[(size_t)r * Cin + ka + 1] : 0.0f;
      if (IN_BN) {
        if (ka < Cin)     a0 = fmaxf(a0 * ing[ka]     + inb[ka],     0.0f);
        if (ka + 1 < Cin) a1 = fmaxf(a1 * ing[ka + 1] + inb[ka + 1], 0.0f);
      }
      float b0 = (cOK && ka < Cin)     ? W[(size_t)ka * Cout + c]       : 0.0f;
      float b1 = (cOK && ka + 1 < Cin) ? W[(size_t)(ka + 1) * Cout + c] : 0.0f;
      v2f av; av.x = a0; av.y = a1;
      v2f bv; bv.x = b0; bv.y = b1;
      acc = __builtin_amdgcn_wmma_f32_16x16x4_f32(false, av, false, bv,
                                                  (short)0, acc, false, false);
    }
  }

  if (c < Cout) {
    for (int i = 0; i < 8; ++i) {
      int orow = row0 + (hl << 3) + i;
      if (orow >= rows) continue;
      float v = acc[i];
      if (outg)      v = v * outg[c] + outb[c];
      else if (outb) v = v + outb[c];
      if (res)       v += res[(size_t)orow * Cout + c];
      if (outRelu)   v = fmaxf(v, 0.0f);
      if (tN > 0) {
        int bb = orow / tN, nn = orow % tN;
        Y[((size_t)bb * Cout + c) * tN + nn] = v;
      } else {
        Y[(size_t)orow * Cout + c] = v;
      }
    }
  }
}

static void linear(hipStream_t s, const float* X, const float* W, float* Y,
                   int rows, int Cin, int Cout,
                   const float* ing, const float* inb,
                   const float* outg, const float* outb, int outRelu,
                   const float* res, int tN)
{
  long long tiles = (long long)((Cout + 15) / 16) * ((rows + 15) / 16);
  int blk = (int)((tiles + 7) / 8);
  if (ing)
    k_linear_wmma<true><<<blk, 256, 0, s>>>(X, W, Y, rows, Cin, Cout,
                                            ing, inb, outg, outb, outRelu, res, tN);
  else
    k_linear_wmma<false><<<blk, 256, 0, s>>>(X, W, Y, rows, Cin, Cout,
                                             nullptr, nullptr, outg, outb, outRelu, res, tN);
}

// ================= elementwise BN(+ReLU), in place =================
__global__ __launch_bounds__(256) void k_bnrelu(float* __restrict__ x,
    const float* __restrict__ g, const float* __restrict__ b, long long rows, int C)
{
  long long id = (long long)blockIdx.x * blockDim.x + threadIdx.x;
  if (id >= rows * C) return;
  int c = (int)(id % C);
  float v = x[id] * g[c] + b[c];
  x[id] = fmaxf(v, 0.0f);
}

// ================= split pc -> p, feat =================
__global__ __launch_bounds__(256) void k_split(const float* __restrict__ pc,
    float* __restrict__ p, float* __restrict__ x, int N)
{
  int id = blockIdx.x * blockDim.x + threadIdx.x;
  if (id >= BB * N) return;
  const float* src = pc + (size_t)id * 9;
  p[(size_t)id * 3 + 0] = src[0];
  p[(size_t)id * 3 + 1] = src[1];
  p[(size_t)id * 3 + 2] = src[2];
  for (int c = 0; c < 6; ++c) x[(size_t)id * 6 + c] = src[3 + c];
}

// ================= brute-force KNN (k smallest sq-dists, stable ties) ========
// When M % 256 == 0 every block serves a single batch, so reference points are
// staged cooperatively through LDS (3KB) and swept via conflict-free broadcasts.
template <int K>
__global__ __launch_bounds__(256) void k_knn(const float* __restrict__ q,
    const float* __restrict__ r, int* __restrict__ idx, float* __restrict__ dst,
    int M, int Mr)
{
  __shared__ float sp[256 * 3];
  int t = blockIdx.x * 256 + threadIdx.x;
  bool valid = t < BB * M;
  int tq = valid ? t : BB * M - 1;
  int b = tq / M;
  float qx = q[(size_t)tq * 3 + 0];
  float qy = q[(size_t)tq * 3 + 1];
  float qz = q[(size_t)tq * 3 + 2];
  float bd[K]; int bi[K];
#pragma unroll
  for (int i = 0; i < K; ++i) { bd[i] = 3.4e38f; bi[i] = 0; }
  const float* rp = r + (size_t)b * Mr * 3;

  if ((M & 255) == 0) {
    for (int c0 = 0; c0 < Mr; c0 += 256) {
      int nc = Mr - c0; if (nc > 256) nc = 256;
      __syncthreads();
      if ((int)threadIdx.x < nc) {
        sp[threadIdx.x * 3 + 0] = rp[(size_t)(c0 + threadIdx.x) * 3 + 0];
        sp[threadIdx.x * 3 + 1] = rp[(size_t)(c0 + threadIdx.x) * 3 + 1];
        sp[threadIdx.x * 3 + 2] = rp[(size_t)(c0 + threadIdx.x) * 3 + 2];
      }
      __syncthreads();
      for (int j = 0; j < nc; ++j) {
        float dx = qx - sp[j * 3 + 0];
        float dy = qy - sp[j * 3 + 1];
        float dz = qz - sp[j * 3 + 2];
        float d = dx * dx + dy * dy + dz * dz;
        if (d < bd[K - 1]) {
          int pos = K - 1;
#pragma unroll
          for (int u = K - 1; u > 0; --u) {
            if (bd[pos - 1] > d) { bd[pos] = bd[pos - 1]; bi[pos] = bi[pos - 1]; pos--; }
          }
          bd[pos] = d; bi[pos] = c0 + j;
        }
      }
    }
  } else {
    for (int j = 0; j < Mr; ++j) {
      if ((j & 63) == 0 && j + 64 < Mr) __builtin_prefetch(rp + (j + 64) * 3, 0, 3);
      float dx = qx - rp[j * 3 + 0];
      float dy = qy - rp[j * 3 + 1];
      float dz = qz - rp[j * 3 + 2];
      float d = dx * dx + dy * dy + dz * dz;
      if (d < bd[K - 1]) {
        int pos = K - 1;
#pragma unroll
        for (int u = K - 1; u > 0; --u) {
          if (bd[pos - 1] > d) { bd[pos] = bd[pos - 1]; bi[pos] = bi[pos - 1]; pos--; }
        }
        bd[pos] = d; bi[pos] = j;
      }
    }
  }

  if (valid) {
#pragma unroll
    for (int kk = 0; kk < K; ++kk) {
      idx[(size_t)t * K + kk] = bi[kk];
      if (dst) dst[(size_t)t * K + kk] = bd[kk];
    }
  }
}

// ================= farthest point sampling (1 workgroup / batch) =============
__global__ __launch_bounds__(256) void k_fps(const float* __restrict__ p,
    int* __restrict__ outIdx, float* __restrict__ dmin, int Np, int m)
{
  int b = blockIdx.x;
  const float* pb = p + (size_t)b * Np * 3;
  float* db = dmin + (size_t)b * Np;
  int* ob = outIdx + (size_t)b * m;
  __shared__ float sv[256];
  __shared__ int si[256];
  __shared__ int s_last;
  int tid = threadIdx.x;
  for (int i = tid; i < Np; i += 256) db[i] = 1.0e10f;
  if (tid == 0) { ob[0] = 0; s_last = 0; }
  __syncthreads();
  for (int j = 1; j < m; ++j) {
    int last = s_last;
    float lx = pb[last * 3 + 0], ly = pb[last * 3 + 1], lz = pb[last * 3 + 2];
    float bv = -1.0f; int bi = 0x7fffffff;
    for (int i = tid; i < Np; i += 256) {
      float dx = pb[i * 3 + 0] - lx;
      float dy = pb[i * 3 + 1] - ly;
      float dz = pb[i * 3 + 2] - lz;
      float d = fminf(db[i], dx * dx + dy * dy + dz * dz);
      db[i] = d;
      if (d > bv) { bv = d; bi = i; }
    }
    sv[tid] = bv; si[tid] = bi;
    __syncthreads();
    for (int s2 = 128; s2 > 0; s2 >>= 1) {
      if (tid < s2) {
        if (sv[tid + s2] > sv[tid] ||
            (sv[tid + s2] == sv[tid] && si[tid + s2] < si[tid])) {
          sv[tid] = sv[tid + s2]; si[tid] = si[tid + s2];
        }
      }
      __syncthreads();
    }
    if (tid == 0) { ob[j] = si[0]; s_last = si[0]; }
    __syncthreads();
  }
}

// ================= gather 3D points by index =================
__global__ __launch_bounds__(256) void k_gather3(const float* __restrict__ p,
    const int* __restrict__ fidx, float* __restrict__ pn, int N, int M)
{
  int id = blockIdx.x * blockDim.x + threadIdx.x;
  if (id >= BB * M) return;
  int b = id / M;
  int j = fidx[id];
  const float* src = p + ((size_t)b * N + j) * 3;
  pn[(size_t)id * 3 + 0] = src[0];
  pn[(size_t)id * 3 + 1] = src[1];
  pn[(size_t)id * 3 + 2] = src[2];
}

// ================= pe step 1: t3 = relu(bn(rel @ p1)) =================
__global__ __launch_bounds__(256) void k_pe1(const float* __restrict__ p,
    const int* __restrict__ idx, const float* __restrict__ p1w,
    const float* __restrict__ pg, const float* __restrict__ pbv,
    float* __restrict__ t3, int N, int K)
{
  long long id = (long long)blockIdx.x * blockDim.x + threadIdx.x;
  long long tot = (long long)BB * N * K;
  if (id >= tot) return;
  long long t = id;
  int n = (int)((t / K) % N);
  int b = (int)(t / ((long long)K * N));
  int j = idx[t];
  float rx = p[((size_t)b * N + j) * 3 + 0] - p[((size_t)b * N + n) * 3 + 0];
  float ry = p[((size_t)b * N + j) * 3 + 1] - p[((size_t)b * N + n) * 3 + 1];
  float rz = p[((size_t)b * N + j) * 3 + 2] - p[((size_t)b * N + n) * 3 + 2];
  for (int c = 0; c < 3; ++c) {
    float v = rx * p1w[0 * 3 + c] + ry * p1w[1 * 3 + c] + rz * p1w[2 * 3 + c];
    v = v * pg[c] + pbv[c];
    t3[t * 3 + c] = fmaxf(v, 0.0f);
  }
}

// ================= pe step 2: pe = t3 @ p2  [tot x C] =================
__global__ __launch_bounds__(256) void k_pe2(const float* __restrict__ t3,
    const float* __restrict__ p2, float* __restrict__ pe, long long tot, int C)
{
  long long id = (long long)blockIdx.x * blockDim.x + threadIdx.x;
  if (id >= tot * C) return;
  long long t = id / C;
  int c = (int)(id % C);
  pe[id] = t3[t * 3 + 0] * p2[0 * C + c] + t3[t * 3 + 1] * p2[1 * C + c] +
           t3[t * 3 + 2] * p2[2 * C + c];
}

// ================= w = gather(k) - q + pe =================
__global__ __launch_bounds__(256) void k_wform(const float* __restrict__ kb,
    const float* __restrict__ qb, const float* __restrict__ pe,
    const int* __restrict__ idx, float* __restrict__ wout, int N, int K, int C)
{
  long long id = (long long)blockIdx.x * blockDim.x + threadIdx.x;
  long long tot = (long long)BB * N * K * C;
  if (id >= tot) return;
  long long t = id / C;
  int c = (int)(id % C);
  int n = (int)((t / K) % N);
  int b = (int)(t / ((long long)K * N));
  int j = idx[t];
  wout[id] = kb[((size_t)b * N + j) * C + c] - qb[((size_t)b * N + n) * C + c] + pe[id];
}

// ================= val = gather(v) + pe  (in-place on pe) =================
__global__ __launch_bounds__(256) void k_valform(const float* __restrict__ vb,
    float* __restrict__ pe, const int* __restrict__ idx, int N, int K, int C)
{
  long long id = (long long)blockIdx.x * blockDim.x + threadIdx.x;
  long long tot = (long long)BB * N * K * C;
  if (id >= tot) return;
  long long t = id / C;
  int c = (int)(id % C);
  int b = (int)(t / ((long long)K * N));
  int j = idx[t];
  pe[id] = vb[((size_t)b * N + j) * C + c] + pe[id];
}

// ================= softmax over K (in-place) =================
template <int K>
__global__ __launch_bounds__(256) void k_softmaxK(float* __restrict__ w,
    long long rows, int Cs)
{
  long long id = (long long)blockIdx.x * blockDim.x + threadIdx.x;
  if (id >= rows * Cs) return;
  long long t = id / Cs;
  int cs = (int)(id % Cs);
  size_t base = (size_t)t * K * Cs + cs;
  float mx = -3.4e38f;
#pragma unroll
  for (int k = 0; k < K; ++k) mx = fmaxf(mx, w[base + (size_t)k * Cs]);
  float sum = 0.0f;
  float e[K];
#pragma unroll
  for (int k = 0; k < K; ++k) { e[k] = __expf(w[base + (size_t)k * Cs] - mx); sum += e[k]; }
  float inv = 1.0f / sum;
#pragma unroll
  for (int k = 0; k < K; ++k) w[base + (size_t)k * Cs] = e[k] * inv;
}

// ================= grouped aggregation einsum =================
__global__ __launch_bounds__(256) void k_agg(const float* __restrict__ val,
    const float* __restrict__ a, float* __restrict__ y, long long rows, int K, int C)
{
  long long id = (long long)blockIdx.x * blockDim.x + threadIdx.x;
  if (id >= rows * C) return;
  long long t = id / C;
  int c = (int)(id % C);
  int Cs = C >> 3;
  int cp = c & (Cs - 1);
  float acc = 0.0f;
  for (int k = 0; k < K; ++k)
    acc += val[((size_t)t * K + k) * C + c] * a[((size_t)t * K + k) * Cs + cp];
  y[id] = acc;
}

// ================= transition-down feature build =================
__global__ __launch_bounds__(256) void k_featdown(const float* __restrict__ p,
    const float* __restrict__ pn, const float* __restrict__ x,
    const int* __restrict__ idx, float* __restrict__ feat,
    int Nref, int M, int K, int C)
{
  long long id = (long long)blockIdx.x * blockDim.x + threadIdx.x;
  long long tot = (long long)BB * M * K;
  if (id >= tot) return;
  long long t = id;
  long long bm = t / K;
  int b = (int)(bm / M);
  int j = idx[t];
  float* f = feat + (size_t)t * (3 + C);
  f[0] = p[((size_t)b * Nref + j) * 3 + 0] - pn[bm * 3 + 0];
  f[1] = p[((size_t)b * Nref + j) * 3 + 1] - pn[bm * 3 + 1];
  f[2] = p[((size_t)b * Nref + j) * 3 + 2] - pn[bm * 3 + 2];
  const float* xs = x + ((size_t)b * Nref + j) * C;
  for (int c = 0; c < C; ++c) f[3 + c] = xs[c];
}

// ================= max over K =================
__global__ __launch_bounds__(256) void k_maxk(const float* __restrict__ y,
    float* __restrict__ out, int M, int K, int C)
{
  long long id = (long long)blockIdx.x * blockDim.x + threadIdx.x;
  if (id >= (long long)BB * M * C) return;
  long long bm = id / C;
  int c = (int)(id % C);
  float mx = -3.4e38f;
  for (int k = 0; k < K; ++k) mx = fmaxf(mx, y[((size_t)bm * K + k) * C + c]);
  out[id] = mx;
}

// ================= 3-NN inverse-distance interpolation, added in place =======
__global__ __launch_bounds__(256) void k_interp(float* __restrict__ yf,
    const float* __restrict__ yc, const int* __restrict__ idx,
    const float* __restrict__ dst, int Nf, int Nc, int C)
{
  long long id = (long long)blockIdx.x * blockDim.x + threadIdx.x;
  if (id >= (long long)BB * Nf * C) return;
  long long bn = id / C;
  int c = (int)(id % C);
  int b = (int)(bn / Nf);
  float w0 = 1.0f / (dst[bn * 3 + 0] + 1e-8f);
  float w1 = 1.0f / (dst[bn * 3 + 1] + 1e-8f);
  float w2 = 1.0f / (dst[bn * 3 + 2] + 1e-8f);
  float inv = 1.0f / (w0 + w1 + w2);
  int i0 = idx[bn * 3 + 0], i1 = idx[bn * 3 + 1], i2 = idx[bn * 3 + 2];
  float acc = w0 * yc[((size_t)b * Nc + i0) * C + c] +
              w1 * yc[((size_t)b * Nc + i1) * C + c] +
              w2 * yc[((size_t)b * Nc + i2) * C + c];
  yf[id] += acc * inv;
}

// ================= host-side parameter plumbing =================
struct Cur { void* const* a; int i; const float* f() { return (const float*)a[i++]; } };
struct TrP { const float *wq,*wk,*wv,*p1,*pg,*pb,*p2,*wg1,*wb1,*wl1,*wg2,*wb2,*wl2; };
struct BlockP { const float *w1,*g1,*b1; TrP tr; const float *g2,*b2,*w3,*g3,*b3; };
struct MlpP { const float *w1,*g1,*b1,*w2,*g2,*b2; };
struct DownP { const float *w,*g,*b; };
struct UpP { const float *ws,*gs,*bs,*wc,*gc,*bc; };
struct OutP { const float *w1,*g1,*b1,*w2,*b2; };

static void readBlock(Cur& c, BlockP& bp) {
  bp.w1=c.f(); bp.g1=c.f(); bp.b1=c.f();
  bp.tr.wq=c.f(); bp.tr.wk=c.f(); bp.tr.wv=c.f();
  bp.tr.p1=c.f(); bp.tr.pg=c.f(); bp.tr.pb=c.f(); bp.tr.p2=c.f();
  bp.tr.wg1=c.f(); bp.tr.wb1=c.f(); bp.tr.wl1=c.f();
  bp.tr.wg2=c.f(); bp.tr.wb2=c.f(); bp.tr.wl2=c.f();
  bp.g2=c.f(); bp.b2=c.f(); bp.w3=c.f(); bp.g3=c.f(); bp.b3=c.f();
}

struct WS {
  char* base; size_t off;
  float* f(size_t n) { float* p=(float*)(base+off); off += ((n*4+255)/256)*256; return p; }
  int*   i(size_t n) { int*   p=(int*)  (base+off); off += ((n*4+255)/256)*256; return p; }
};

struct Bufs {
  float *p[5], *x[5], *yd[5];
  int *idx[5];
  float *p1coords, *x0, *tmpA, *tmpB, *qb, *kb, *vb;
  float *t3, *pe, *wb, *w2b, *w3b;
  float *feat, *ydown, *fpsD, *ycB, *distU;
  int *fpsI, *idxDown, *idxU;
};

static void knn_dispatch(hipStream_t s, const float* q, const float* r, int* idx,
                         float* dst, int M, int Mr, int K)
{
  int g = nblocks((long long)BB * M);
  if (K == 3)       k_knn<3><<<g,256,0,s>>>(q, r, idx, dst, M, Mr);
  else if (K == 8)  k_knn<8><<<g,256,0,s>>>(q, r, idx, dst, M, Mr);
  else              k_knn<16><<<g,256,0,s>>>(q, r, idx, dst, M, Mr);
}

static void run_block(hipStream_t s, const BlockP& bp, const float* p, float* x,
                      const int* idx, int N, int C, int K, Bufs& B)
{
  int rows = BB * N;
  long long tot = (long long)rows * K;
  int Cs = C / 8;
  // y = relu(bn(x @ w1))
  linear(s, x, bp.w1, B.tmpA, rows, C, C, 0,0, bp.g1, bp.b1, 1, 0, 0);
  // q,k,v
  linear(s, B.tmpA, bp.tr.wq, B.qb, rows, C, C, 0,0, 0,0,0, 0, 0);
  linear(s, B.tmpA, bp.tr.wk, B.kb, rows, C, C, 0,0, 0,0,0, 0, 0);
  linear(s, B.tmpA, bp.tr.wv, B.vb, rows, C, C, 0,0, 0,0,0, 0, 0);
  // positional encoding
  k_pe1<<<nblocks(tot),256,0,s>>>(p, idx, bp.tr.p1, bp.tr.pg, bp.tr.pb, B.t3, N, K);
  k_pe2<<<nblocks(tot*C),256,0,s>>>(B.t3, bp.tr.p2, B.pe, tot, C);
  // w = kj - q + pe
  k_wform<<<nblocks(tot*C),256,0,s>>>(B.kb, B.qb, B.pe, idx, B.wb, N, K, C);
  // w MLP (input BN+ReLU fused into the GEMM)
  linear(s, B.wb,  bp.tr.wl1, B.w2b, (int)tot, C,  Cs, bp.tr.wg1, bp.tr.wb1, 0,0,0, 0, 0);
  linear(s, B.w2b, bp.tr.wl2, B.w3b, (int)tot, Cs, Cs, bp.tr.wg2, bp.tr.wb2, 0,0,0, 0, 0);
  // softmax over K
  if (K == 8) k_softmaxK<8><<<nblocks((long long)rows*Cs),256,0,s>>>(B.w3b, rows, Cs);
  else        k_softmaxK<16><<<nblocks((long long)rows*Cs),256,0,s>>>(B.w3b, rows, Cs);
  // val = vj + pe (in place), aggregate
  k_valform<<<nblocks(tot*C),256,0,s>>>(B.vb, B.pe, idx, N, K, C);
  k_agg<<<nblocks((long long)rows*C),256,0,s>>>(B.pe, B.w3b, B.tmpB, rows, K, C);
  // y = relu(bn(pt_out)); x = relu(bn(y @ w3) + x)
  k_bnrelu<<<nblocks((long long)rows*C),256,0,s>>>(B.tmpB, bp.g2, bp.b2, rows, C);
  linear(s, B.tmpB, bp.w3, x, rows, C, C, 0,0, bp.g3, bp.b3, 1, x, 0);
}

static void run_stage(hipStream_t s, BlockP* bps, int nb, const float* p, float* x,
                      const int* idx, int N, int C, int K, Bufs& B)
{
  for (int i = 0; i < nb; ++i) run_block(s, bps[i], p, x, idx, N, C, K, B);
}

static void run_down(hipStream_t s, const DownP& dp, const float* pIn, const float* xIn,
                     int Nin, int Cin, float* pOut, float* xOut, int M, int Cout, Bufs& B)
{
  const int K = 16;
  k_fps<<<BB,256,0,s>>>(pIn, B.fpsI, B.fpsD, Nin, M);
  k_gather3<<<nblocks((long long)BB*M),256,0,s>>>(pIn, B.fpsI, pOut, Nin, M);
  knn_dispatch(s, pOut, pIn, B.idxDown, nullptr, M, Nin, K);
  k_featdown<<<nblocks((long long)BB*M*K),256,0,s>>>(pIn, pOut, xIn, B.idxDown,
                                                     B.feat, Nin, M, K, Cin);
  linear(s, B.feat, dp.w, B.ydown, BB*M*K, 3+Cin, Cout, 0,0, dp.g, dp.b, 1, 0, 0);
  k_maxk<<<nblocks((long long)BB*M*Cout),256,0,s>>>(B.ydown, xOut, M, K, Cout);
}

static void run_up(hipStream_t s, const UpP& up, const float* pc_, const float* yc,
                   int Nc, int Cc, const float* pf, const float* xf, int Nf, int Cf,
                   float* out, Bufs& B)
{
  linear(s, xf, up.ws, out,   BB*Nf, Cf, Cf, 0,0, up.gs, up.bs, 1, 0, 0);
  linear(s, yc, up.wc, B.ycB, BB*Nc, Cc, Cf, 0,0, up.gc, up.bc, 1, 0, 0);
  knn_dispatch(s, pf, pc_, B.idxU, B.distU, Nf, Nc, 3);
  k_interp<<<nblocks((long long)BB*Nf*Cf),256,0,s>>>(out, B.ycB, B.idxU, B.distU, Nf, Nc, Cf);
}

// ================= entry point =================
extern "C" void kernel_launch(void* const* d_in, const int* in_sizes, int n_in,
                              void* d_out, int out_size, void* d_ws, size_t ws_size,
                              hipStream_t stream)
{
  Cur cur{d_in, 0};
  const float* pc = cur.f();
  MlpP im; im.w1=cur.f(); im.g1=cur.f(); im.b1=cur.f(); im.w2=cur.f(); im.g2=cur.f(); im.b2=cur.f();
  static BlockP enc[5][6];
  for (int i = 0; i < 5; ++i)
    for (int j = 0; j < NBLK[i]; ++j) readBlock(cur, enc[i][j]);
  DownP down[4];
  for (int i = 0; i < 4; ++i) { down[i].w=cur.f(); down[i].g=cur.f(); down[i].b=cur.f(); }
  MlpP dm; dm.w1=cur.f(); dm.g1=cur.f(); dm.b1=cur.f(); dm.w2=cur.f(); dm.g2=cur.f(); dm.b2=cur.f();
  static BlockP dec[5][2];
  for (int i = 0; i < 5; ++i)
    for (int j = 0; j < 2; ++j) readBlock(cur, dec[i][j]);
  UpP up[4];
  for (int i = 0; i < 4; ++i) { up[i].ws=cur.f(); up[i].gs=cur.f(); up[i].bs=cur.f();
                                up[i].wc=cur.f(); up[i].gc=cur.f(); up[i].bc=cur.f(); }
  OutP om; om.w1=cur.f(); om.g1=cur.f(); om.b1=cur.f(); om.w2=cur.f(); om.b2=cur.f();

  // ---- workspace layout (deterministic) ----
  WS ws{(char*)d_ws, 0};
  Bufs B;
  B.p1coords = ws.f((size_t)BB*NL[0]*3);  B.p[0] = B.p1coords;
  for (int l = 1; l < 5; ++l) B.p[l] = ws.f((size_t)BB*NL[l]*3);
  B.x0 = ws.f((size_t)BB*NL[0]*6);
  for (int l = 0; l < 5; ++l) B.x[l]  = ws.f((size_t)BB*NL[l]*CL[l]);
  for (int l = 0; l < 5; ++l) B.yd[l] = ws.f((size_t)BB*NL[l]*CL[l]);
  for (int l = 0; l < 5; ++l) B.idx[l] = ws.i((size_t)BB*NL[l]*KL[l]);
  B.tmpA = ws.f(524288);  B.tmpB = ws.f(524288);
  B.qb = ws.f(524288);    B.kb = ws.f(524288);   B.vb = ws.f(524288);
  B.t3 = ws.f(393216);
  B.pe = ws.f(4194304);   B.wb = ws.f(4194304);
  B.w2b = ws.f(524288);   B.w3b = ws.f(524288);
  B.feat = ws.f(2293760); B.ydown = ws.f(4194304);
  B.fpsD = ws.f((size_t)BB*NL[0]);  B.fpsI = ws.i((size_t)BB*NL[1]);
  B.idxDown = ws.i((size_t)BB*NL[1]*16);
  B.ycB = ws.f(131072);
  B.idxU = ws.i((size_t)BB*NL[0]*3); B.distU = ws.f((size_t)BB*NL[0]*3);
  (void)ws_size; (void)in_sizes; (void)n_in; (void)out_size;

  // ---- encoder ----
  k_split<<<nblocks((long long)BB*NL[0]),256,0,stream>>>(pc, B.p[0], B.x0, NL[0]);
  linear(stream, B.x0,  im.w1, B.tmpA, BB*NL[0], 6,  32, 0,0, im.g1, im.b1, 1, 0, 0);
  linear(stream, B.tmpA, im.w2, B.x[0], BB*NL[0], 32, 32, 0,0, im.g2, im.b2, 1, 0, 0);

  knn_dispatch(stream, B.p[0], B.p[0], B.idx[0], nullptr, NL[0], NL[0], KL[0]);
  run_stage(stream, enc[0], NBLK[0], B.p[0], B.x[0], B.idx[0], NL[0], CL[0], KL[0], B);
  for (int l = 0; l < 4; ++l) {
    run_down(stream, down[l], B.p[l], B.x[l], NL[l], CL[l], B.p[l+1], B.x[l+1], NL[l+1], CL[l+1], B);
    knn_dispatch(stream, B.p[l+1], B.p[l+1], B.idx[l+1], nullptr, NL[l+1], NL[l+1], KL[l+1]);
    run_stage(stream, enc[l+1], NBLK[l+1], B.p[l+1], B.x[l+1], B.idx[l+1], NL[l+1], CL[l+1], KL[l+1], B);
  }

  // ---- decoder ----
  linear(stream, B.x[4], dm.w1, B.tmpA, BB*NL[4], CL[4], CL[4], 0,0, dm.g1, dm.b1, 1, 0, 0);
  linear(stream, B.tmpA, dm.w2, B.yd[4], BB*NL[4], CL[4], CL[4], 0,0, dm.g2, dm.b2, 1, 0, 0);
  run_stage(stream, dec[0], 2, B.p[4], B.yd[4], B.idx[4], NL[4], CL[4], KL[4], B);
  for (int u = 0; u < 4; ++u) {
    int lc = 4 - u, lf = 3 - u;   // coarse, fine levels
    run_up(stream, up[u], B.p[lc], B.yd[lc], NL[lc], CL[lc],
           B.p[lf], B.x[lf], NL[lf], CL[lf], B.yd[lf], B);
    run_stage(stream, dec[u+1], 2, B.p[lf], B.yd[lf], B.idx[lf], NL[lf], CL[lf], KL[lf], B);
  }

  // ---- head: relu(bn(y@w1)); (y@w2 + b2) transposed to (B, NCLS, N) ----
  linear(stream, B.yd[0], om.w1, B.tmpA, BB*NL[0], CL[0], CL[0], 0,0, om.g1, om.b1, 1, 0, 0);
  linear(stream, B.tmpA, om.w2, (float*)d_out, BB*NL[0], CL[0], NCLS,
         0,0, nullptr, om.b2, 0, 0, NL[0]);
}